// HistogramLoss_50620484551215
// MI455X (gfx1250) — compile-verified
//
#include <hip/hip_runtime.h>
#include <stdint.h>

// ---------------------------------------------------------------------------
// HistogramLoss for MI455X (gfx1250, wave32).
// input  [8,64,256,256] f32  -> 512 channels x 65536
// target [8,64,128,128] f32  -> 512 channels x 16384
// out = matched (512*65536 f32) ++ loss (1 f32)
//
// One workgroup (1024 threads, 32 waves) per channel.
// All sorting is LDS-resident (8-pass 4-bit LSD radix, ballot-based stable
// ranking). Sorted target spills to d_ws (L2-resident). Async LDS loads +
// WMMA reduction exercise CDNA5-specific instruction paths.
// ---------------------------------------------------------------------------

#define BC       512
#define S_ELEMS  65536
#define R_ELEMS  16384
#define NTH      1024
#define CHUNK    16384
#define NCHUNK   4

// LDS layout (dword indices within the static shared block)
#define LDS_A_OFF    0                  // 16384 u32 : sort ping
#define LDS_B_OFF    16384              // 16384 u32 : sort pong
#define LDS_RANK_OFF 32768              // 32768 u32 : 65536 u16 ranks
#define LDS_RED_OFF  (32768 + 32768)    //    32 f32 : per-wave reduction
#define SMEM_DWORDS  (16384 + 16384 + 32768 + 32)   // 262272 bytes total

typedef __attribute__((ext_vector_type(2))) float v2f;
typedef __attribute__((ext_vector_type(8))) float v8f;

// Order-preserving float->uint key transform (IEEE total order, NaN-free data)
__device__ __forceinline__ unsigned fkey_from_bits(unsigned u) {
  return u ^ ((unsigned)((int)u >> 31) | 0x80000000u);
}
__device__ __forceinline__ float inv_fkey(unsigned k) {
  unsigned u = (k & 0x80000000u) ? (k ^ 0x80000000u) : ~k;
  return __uint_as_float(u);
}

// CDNA5 async global->LDS copy of one dword per lane.
// Per ISA flat aperture rules, the low 32 bits of a flat LDS address are the
// wave-relative LDS byte offset, which is what the VDST operand expects.
__device__ __forceinline__ void async_load_dword(const float* __restrict__ g,
                                                 const unsigned* lds_dst) {
  unsigned lds_byte = (unsigned)(uintptr_t)(const void*)lds_dst;
  asm volatile("global_load_async_to_lds_b32 %0, %1, off"
               :: "v"(lds_byte), "v"(g)
               : "memory");
}

// Fill 16384 dwords from global into LDS starting at lds_base.
__device__ __forceinline__ void async_chunk_load(const float* __restrict__ gsrc,
                                                 const unsigned* lds_base) {
  const unsigned t = threadIdx.x;
#pragma unroll
  for (int i = 0; i < 16; ++i) {
    unsigned idx = t + (unsigned)i * NTH;
    async_load_dword(gsrc + idx, lds_base + idx);
  }
  asm volatile("s_wait_asynccnt 0" ::: "memory");
}

// Stable LSD radix sort of 16384 keys held in LDS buffer a (ping) / b (pong).
// Pass 0 reads raw float bits from a and applies the key transform on the fly.
// 8 passes (4-bit digits) => result (transformed keys, ascending) ends in a.
// Cross-wave digit-count scratch overlays dst[0..511] (dead until scatter).
__device__ void sort16k(unsigned* __restrict__ a, unsigned* __restrict__ b) {
  const unsigned tid = threadIdx.x;
  const unsigned wv  = tid >> 5;          // wave 0..31
  const unsigned ln  = tid & 31u;         // lane 0..31
  const unsigned ltm = (1u << ln) - 1u;   // lanes-below mask
  unsigned* src = a;
  unsigned* dst = b;
#pragma unroll 1
  for (int pass = 0; pass < 8; ++pass) {
    const unsigned shift = (unsigned)pass * 4u;
    unsigned keys[16];
    unsigned meta[16];                    // (digit<<16) | rank-in-wave
    unsigned rc = 0;                      // lane l<16: running count of digit l
    const unsigned base = wv * 512u;
#pragma unroll
    for (int i = 0; i < 16; ++i) {
      unsigned raw = src[base + (unsigned)i * 32u + ln];
      unsigned k = (pass == 0) ? fkey_from_bits(raw) : raw;
      unsigned d = (k >> shift) & 15u;
      unsigned b0 = __builtin_amdgcn_ballot_w32((d & 1u) != 0u);
      unsigned b1 = __builtin_amdgcn_ballot_w32((d & 2u) != 0u);
      unsigned b2 = __builtin_amdgcn_ballot_w32((d & 4u) != 0u);
      unsigned b3 = __builtin_amdgcn_ballot_w32((d & 8u) != 0u);
      unsigned pm = ((d & 1u) ? b0 : ~b0) & ((d & 2u) ? b1 : ~b1) &
                    ((d & 4u) ? b2 : ~b2) & ((d & 8u) ? b3 : ~b3);
      unsigned prefix =
          (unsigned)__builtin_amdgcn_ds_bpermute((int)(d << 2), (int)rc);
      unsigned wrank = prefix + (unsigned)__popc(pm & ltm);
      keys[i] = k;
      meta[i] = (d << 16) | wrank;
      // per-iteration count of digit==lane (ballots are wave-uniform)
      unsigned lm = ((ln & 1u) ? b0 : ~b0) & ((ln & 2u) ? b1 : ~b1) &
                    ((ln & 4u) ? b2 : ~b2) & ((ln & 8u) ? b3 : ~b3);
      rc += (unsigned)__popc(lm);
    }
    if (ln < 16u) dst[wv * 16u + ln] = rc;    // publish per-wave digit counts
    __syncthreads();
    // each wave: exclusive bases. dl = digit handled by this lane.
    const unsigned dl = ln & 15u;
    unsigned pre = 0, tot = 0;
    for (unsigned w2 = 0; w2 < 32u; ++w2) {
      unsigned v = dst[w2 * 16u + dl];
      pre += (w2 < wv) ? v : 0u;
      tot += v;
    }
    unsigned sc = tot;                        // inclusive scan over digits
#pragma unroll
    for (unsigned off = 1; off < 16u; off <<= 1) {
      unsigned nb = (unsigned)__builtin_amdgcn_ds_bpermute(
          (int)(((ln - off) & 31u) << 2), (int)sc);
      sc += (dl >= off) ? nb : 0u;
    }
    unsigned mybase = (sc - tot) + pre;       // valid on lanes 0..15
    __syncthreads();                          // scratch fully consumed
#pragma unroll
    for (int i = 0; i < 16; ++i) {
      unsigned d = meta[i] >> 16;
      unsigned bpos =
          (unsigned)__builtin_amdgcn_ds_bpermute((int)(d << 2), (int)mybase);
      dst[bpos + (meta[i] & 0xFFFFu)] = keys[i];
    }
    __syncthreads();
    unsigned* tmp = src; src = dst; dst = tmp;
  }
}

// count of elements < k in a sorted(uint-key) LDS array of 16384
__device__ __forceinline__ unsigned lb16k(const unsigned* __restrict__ a,
                                          unsigned k) {
  unsigned pos = 0;
#pragma unroll
  for (int s = 13; s >= 0; --s) {
    unsigned cand = pos + (1u << s);
    unsigned v = a[cand - 1u];
    pos = (v < k) ? cand : pos;
  }
  return pos;
}

__global__ __launch_bounds__(NTH) void hist_match_kernel(
    const float* __restrict__ in, const float* __restrict__ tgt,
    float* __restrict__ matched, float* __restrict__ ws_ref,
    float* __restrict__ ws_partial) {
  __shared__ unsigned smem[SMEM_DWORDS];    // 262272 B static LDS
  unsigned* bufA = smem + LDS_A_OFF;
  unsigned* bufB = smem + LDS_B_OFF;
  unsigned short* ranks = (unsigned short*)(smem + LDS_RANK_OFF);
  float* red = (float*)(smem + LDS_RED_OFF);

  const unsigned bc  = blockIdx.x;
  const unsigned tid = threadIdx.x;
  const float* chan  = in  + (size_t)bc * S_ELEMS;
  const float* tchan = tgt + (size_t)bc * R_ELEMS;
  float* outc = matched + (size_t)bc * S_ELEMS;
  float* refc = ws_ref  + (size_t)bc * R_ELEMS;

  // ---- Stage 1: sort target channel, spill sorted ref (floats) to ws ----
  async_chunk_load(tchan, bufA);
  __syncthreads();
  sort16k(bufA, bufB);
#pragma unroll
  for (int i = 0; i < 16; ++i) {
    unsigned idx = tid + (unsigned)i * NTH;
    refc[idx] = inv_fkey(bufA[idx]);
  }
  __threadfence();
  __syncthreads();

  // ---- Stage 2: per 16K chunk: sort, then accumulate per-element ranks ----
  for (int c = 0; c < NCHUNK; ++c) {
    async_chunk_load(chan + (size_t)c * CHUNK, bufA);
    __syncthreads();
    sort16k(bufA, bufB);
    for (int i = 0; i < 64; ++i) {
      unsigned p = tid + (unsigned)i * NTH;
      unsigned k = fkey_from_bits(__float_as_uint(chan[p]));
      unsigned cnt = lb16k(bufA, k);
      if (c == 0) ranks[p] = (unsigned short)cnt;
      else        ranks[p] = (unsigned short)(ranks[p] + cnt);
    }
    __syncthreads();
  }

  // ---- Stage 3: resample sorted ref at rank positions, write matched ----
  const float stepf = (float)(R_ELEMS - 1) / (float)(S_ELEMS - 1);
  float local = 0.0f;
  for (int i = 0; i < 64; ++i) {
    unsigned p = tid + (unsigned)i * NTH;
    float x = chan[p];
    unsigned r = ranks[p];
    float t1 = (float)r * stepf;
    float fl = floorf(t1);
    float w  = t1 - fl;
    int lo = (int)fl;
    int hi = lo + 1; if (hi > R_ELEMS - 1) hi = R_ELEMS - 1;
    float m = refc[lo] * (1.0f - w) + refc[hi] * w;   // L2-resident gathers
    outc[p] = m;
    float d = x - m;
    local = fmaf(d, d, local);
  }

  // ---- wave reduction via V_WMMA_F32_16X16X4_F32 (B = ones) ----
  // A layout (16x4 f32): VGPR0 lanes0-15 K=0, lanes16-31 K=2; VGPR1 K=1/K=3.
  // With a=(local,0) and B=all-ones: D[m][n] = local[m] + local[m+16].
  v2f av; av.x = local; av.y = 0.0f;
  v2f bo; bo.x = 1.0f;  bo.y = 1.0f;
  v8f cacc = {0.f, 0.f, 0.f, 0.f, 0.f, 0.f, 0.f, 0.f};
  cacc = __builtin_amdgcn_wmma_f32_16x16x4_f32(false, av, false, bo,
                                               (short)0, cacc, false, false);
  float srows = cacc[0] + cacc[1] + cacc[2] + cacc[3] +
                cacc[4] + cacc[5] + cacc[6] + cacc[7];
  // lane 0 holds rows 0..7 (col 0), lane 16 holds rows 8..15 (col 0)
  float wsum = __shfl(srows, 0, 32) + __shfl(srows, 16, 32);

  const unsigned wv = tid >> 5, ln = tid & 31u;
  if (ln == 0) red[wv] = wsum;
  __syncthreads();
  if (tid == 0) {
    float t2 = 0.0f;
    for (int w2 = 0; w2 < 32; ++w2) t2 += red[w2];   // fixed order: determin.
    ws_partial[bc] = t2;
  }
}

__global__ void finalize_kernel(const float* __restrict__ partials,
                                float* __restrict__ loss) {
  if (threadIdx.x == 0 && blockIdx.x == 0) {
    float s = 0.0f;
    for (int i = 0; i < BC; ++i) s += partials[i];   // fixed order
    loss[0] = s / 33554432.0f;                       // mean over 8*64*256*256
  }
}

extern "C" void kernel_launch(void* const* d_in, const int* in_sizes, int n_in,
                              void* d_out, int out_size, void* d_ws,
                              size_t ws_size, hipStream_t stream) {
  (void)in_sizes; (void)n_in; (void)out_size; (void)ws_size;
  const float* in  = (const float*)d_in[0];
  const float* tgt = (const float*)d_in[1];
  float* matched = (float*)d_out;
  float* loss    = matched + (size_t)BC * S_ELEMS;     // out is matched ++ loss
  float* ws_ref  = (float*)d_ws;                        // 512*16384 f32 = 32MB
  float* ws_part = ws_ref + (size_t)BC * R_ELEMS;       // 512 f32

  hist_match_kernel<<<dim3(BC), dim3(NTH), 0, stream>>>(
      in, tgt, matched, ws_ref, ws_part);
  finalize_kernel<<<dim3(1), dim3(32), 0, stream>>>(ws_part, loss);
}